// SSVM_85796266705187
// MI455X (gfx1250) — compile-verified
//
#include <hip/hip_runtime.h>
#include <stdint.h>

#define BB 64
#define LL 512
#define TT 52
#define CHUNK 64            // timesteps per TDM chunk
#define NCHUNK (LL / CHUNK) // 8

typedef unsigned int u32x4 __attribute__((ext_vector_type(4)));
typedef unsigned int u32x8 __attribute__((ext_vector_type(8)));

// Issue one TDM 2D tile load (global -> LDS). Uniform args; wave-level op.
__device__ __forceinline__ void tdm_load_2d(const void* gptr, uint32_t lds_off,
                                            uint32_t tile_d0, uint32_t tile_d1,
                                            uint32_t tensor_d0, uint32_t tensor_d1,
                                            uint32_t stride0)
{
    uint64_t ga = (uint64_t)(uintptr_t)gptr;

    u32x4 g0;
    g0[0] = 1u;                                    // count=1, is_restore=0, gather off
    g0[1] = lds_off;                               // lds_addr        bits [63:32]
    g0[2] = (uint32_t)(ga & 0xFFFFFFFFu);          // global_addr lo  bits [95:64]
    g0[3] = (uint32_t)((ga >> 32) & 0x01FFFFFFu)   // global_addr hi  bits [120:96]
          | (2u << 30);                            // type=2 ("image")

    u32x8 g1;
    g1[0] = (2u << 16);                            // data_size=2 (4 bytes); wg_mask=0
    g1[1] = (tensor_d0 & 0xFFFFu) << 16;           // tensor_dim0 lo16 @ [63:48]
    g1[2] = (tensor_d0 >> 16) | ((tensor_d1 & 0xFFFFu) << 16); // dim0 hi16 | dim1 lo16
    g1[3] = (tensor_d1 >> 16) | (tile_d0 << 16);   // dim1 hi16 | tile_dim0 @ [127:112]
    g1[4] = tile_d1;                               // tile_dim1 @ [143:128]; tile_dim2=0
    g1[5] = stride0;                               // tensor_dim0_stride low32
    g1[6] = 0u;                                    // stride0 hi16 | dim1_stride lo16 (2D: unused)
    g1[7] = 0u;

    asm volatile("tensor_load_to_lds %0, %1" :: "s"(g0), "s"(g1) : "memory");
}

// s_wait_tensorcnt needs a frontend-constant immediate -> dispatch via switch.
__device__ __forceinline__ void wait_tensor_chunks_done(int c)
{
    switch (c) {
        case 1: __builtin_amdgcn_s_wait_tensorcnt(6); break;
        case 2: __builtin_amdgcn_s_wait_tensorcnt(5); break;
        case 3: __builtin_amdgcn_s_wait_tensorcnt(4); break;
        case 4: __builtin_amdgcn_s_wait_tensorcnt(3); break;
        case 5: __builtin_amdgcn_s_wait_tensorcnt(2); break;
        case 6: __builtin_amdgcn_s_wait_tensorcnt(1); break;
        default: __builtin_amdgcn_s_wait_tensorcnt(0); break;
    }
}

__global__ __launch_bounds__(64) void viterbi_cdna5_kernel(
    const float* __restrict__ feats,   // (B, L, T) f32
    const int*   __restrict__ mask,    // (B, L) i32
    const float* __restrict__ trans,   // (T, T) f32
    float*       __restrict__ out)     // (B, L) path ids as f32
{
    __shared__ float             sFeats[LL * TT];         // 106496 B, filled by TDM chunks
    __shared__ float             sTrans[TT * TT];         // 10816 B, filled by TDM
    alignas(16) __shared__ float sPart[2][TT];            // double-buffered partition
    __shared__ float             sLast[TT];               // hist[len-1]
    __shared__ int               sRed[64];
    __shared__ int               sLen;
    __shared__ uint8_t           sBP[(LL - 1) * TT];      // bp for step t stored at row (t-1)

    const int b  = blockIdx.x;
    const int j  = threadIdx.x;
    const int jc = (j < TT) ? j : 0;                      // clamped lane->tag for safe reads
    const size_t base = (size_t)b * LL * TT;

    // ------------- TDM pipeline: 1 transitions load + 8 feats chunks (wave 0) -------------
    if (j < 32) {
        tdm_load_2d(trans,
                    (uint32_t)(uintptr_t)&sTrans[0],
                    /*tile*/ TT, TT, /*tensor*/ TT, TT, /*stride0*/ TT);
        #pragma unroll
        for (int c = 0; c < NCHUNK; ++c) {
            // each chunk is CHUNK*TT contiguous floats -> 1 row of 3328 elements
            tdm_load_2d(feats + base + (size_t)c * CHUNK * TT,
                        (uint32_t)(uintptr_t)&sFeats[c * CHUNK * TT],
                        /*tile*/ CHUNK * TT, 1, /*tensor*/ CHUNK * TT, 1,
                        /*stride0*/ CHUNK * TT);
        }
    }

    // ------------- overlap DMA latency: sequence-length reduction from mask ---------------
    {
        int acc = 0;
        #pragma unroll
        for (int k = 0; k < 8; ++k)
            acc += mask[(size_t)b * LL + j + 64 * k];
        sRed[j] = acc;
    }

    // need transitions + chunk 0: first 2 of 9 in-order TDM ops -> TENSORcnt <= 7
    __builtin_amdgcn_s_wait_tensorcnt(7);
    __syncthreads();

    if (j == 0) {
        int s = 0;
        #pragma unroll 8
        for (int k = 0; k < 64; ++k) s += sRed[k];
        sLen = s;
    }
    __syncthreads();
    const int len = sLen;

    // ------------- hoist this thread's transition column into registers -------------------
    float treg[TT];
    #pragma unroll
    for (int i = 0; i < TT; ++i) treg[i] = sTrans[i * TT + jc];

    // ------------- t = 0 : partition0[j] = feats[b,0,j] + trans[START,j] ------------------
    if (j < TT) {
        float p0 = sFeats[j] + treg[TT - 2];
        sPart[0][j] = p0;
        if (len - 1 == 0) sLast[j] = p0;
    }
    __syncthreads();

    // ------------- forward max-plus scan, chunk-synchronized with the TDM pipeline --------
    int cur = 0;
    for (int c = 0; c < NCHUNK; ++c) {
        if (c > 0) {
            // chunks complete in order: trans + chunks 0..c done -> TENSORcnt <= 7-c
            wait_tensor_chunks_done(c);
            __syncthreads();
        }
        const int tBeg = (c == 0) ? 1 : c * CHUNK;
        const int tEnd = (c + 1) * CHUNK;
        for (int t = tBeg; t < tEnd; ++t) {
            if (j < TT) {
                const float4* pp4 = (const float4*)sPart[cur];   // broadcast reads
                float feat = sFeats[t * TT + j];
                float best = -__builtin_huge_valf();
                int   bi   = 0;
                // 13 groups of 4; tournament preserves first-occurrence argmax:
                // a later index wins only on strict '>'
                #pragma unroll
                for (int g = 0; g < TT / 4; ++g) {
                    const int i0 = 4 * g;
                    float4 p = pp4[g];
                    float v0 = p.x + treg[i0 + 0];
                    float v1 = p.y + treg[i0 + 1];
                    float v2 = p.z + treg[i0 + 2];
                    float v3 = p.w + treg[i0 + 3];
                    float m01 = v0; int i01 = i0;
                    if (v1 > m01) { m01 = v1; i01 = i0 + 1; }
                    float m23 = v2; int i23 = i0 + 2;
                    if (v3 > m23) { m23 = v3; i23 = i0 + 3; }
                    float m = m01; int im = i01;
                    if (m23 > m) { m = m23; im = i23; }
                    if (m > best) { best = m; bi = im; }
                }
                float newp = best + feat;
                sBP[(t - 1) * TT + j] = (uint8_t)((t < len) ? bi : 0); // mask => bp = 0
                sPart[cur ^ 1][j] = newp;
                if (t == len - 1) sLast[j] = newp;                     // capture hist[len-1]
            }
            __syncthreads();
            cur ^= 1;
        }
    }

    // ------------- termination + backtrace (serial, fully LDS-resident) -------------------
    if (j == 0) {
        const int END = TT - 1;
        float bestv = -__builtin_huge_valf();
        int   bp0   = 0;
        for (int i = 0; i < TT; ++i) {
            float v = sLast[i] + sTrans[i * TT + END];
            if (v > bestv) { bestv = v; bp0 = i; }
        }
        float* o = out + (size_t)b * LL;
        o[LL - 1] = (float)bp0;                     // decode[L-1] = pointer0
        int ptr = bp0;
        for (int s = LL - 2; s >= 0; --s) {
            int val = (s == len - 1) ? bp0          // back[len-1] overwritten w/ pointer0
                                     : (int)sBP[s * TT + ptr]; // masked region stores 0
            ptr = val;
            o[s] = (float)val;
        }
    }
}

extern "C" void kernel_launch(void* const* d_in, const int* in_sizes, int n_in,
                              void* d_out, int out_size, void* d_ws, size_t ws_size,
                              hipStream_t stream) {
    const float* feats = (const float*)d_in[0];   // (64, 512, 52) f32
    const int*   mask  = (const int*)  d_in[1];   // (64, 512) i32
    const float* trans = (const float*)d_in[2];   // (52, 52) f32
    float*       out   = (float*)d_out;           // (64, 512)

    (void)in_sizes; (void)n_in; (void)out_size; (void)d_ws; (void)ws_size;

    viterbi_cdna5_kernel<<<BB, 64, 0, stream>>>(feats, mask, trans, out);
}